// SimpleDeepModel_3075196584230
// MI455X (gfx1250) — compile-verified
//
#include <hip/hip_runtime.h>
#include <hip/hip_bf16.h>

typedef __attribute__((ext_vector_type(2))) float v2f;
typedef __attribute__((ext_vector_type(8))) float v8f;

#define HID   100
#define G4    400
#define SEQL  8192
#define NCLS  20

// ---------------------------------------------------------------------------
// Phase 1: x_proj[n, 0:400] = embedding[sequence[n]] @ W_ih^T + b_ih + b_hh
// One wave (32 threads) per 16x16 output tile, f32 WMMA, K=100 in 25 steps of 4.
// ---------------------------------------------------------------------------
__global__ void __launch_bounds__(32) xproj_wmma_kernel(
    const int*   __restrict__ seq,
    const float* __restrict__ emb,
    const float* __restrict__ W_ih,
    const float* __restrict__ b_ih,
    const float* __restrict__ b_hh,
    float*       __restrict__ xproj)
{
    const int mt   = blockIdx.x;        // 0..511  (M tile: timesteps)
    const int nt   = blockIdx.y;        // 0..24   (N tile: gate rows)
    const int lane = threadIdx.x;       // 0..31
    const int half = lane >> 4;         // 0: lanes 0-15, 1: lanes 16-31
    const int ln   = lane & 15;

    const int  m    = mt * 16 + ln;                 // timestep (A row)
    const int  n    = nt * 16 + ln;                 // gate row (B col)
    const long arow = (long)seq[m] * HID;           // embedding row base
    const long brow = (long)n * HID;                // W_ih row base

    v8f c = {};
    #pragma unroll
    for (int k0 = 0; k0 < HID; k0 += 4) {
        // A 16x4 layout: VGPR0 = K=k0 (lanes 0-15) / K=k0+2 (lanes 16-31),
        //                VGPR1 = K=k0+1           / K=k0+3
        const int ka = k0 + (half ? 2 : 0);
        v2f a, b;
        a.x = emb[arow + ka];
        a.y = emb[arow + ka + 1];
        // B 4x16: row K striped across lanes; same half-split as A.
        b.x = W_ih[brow + ka];
        b.y = W_ih[brow + ka + 1];
        c = __builtin_amdgcn_wmma_f32_16x16x4_f32(
                /*neg_a=*/false, a, /*neg_b=*/false, b,
                /*c_mod=*/(short)0, c, /*reuse_a=*/false, /*reuse_b=*/false);
    }

    // Bias is per output column n (same for all 8 rows this lane holds).
    const float bias = b_ih[n] + b_hh[n];

    // C/D layout: VGPR v, lanes 0-15 -> M = v, lanes 16-31 -> M = v + 8.
    #pragma unroll
    for (int v = 0; v < 8; ++v) {
        const int row = mt * 16 + v + half * 8;
        xproj[(long)row * G4 + nt * 16 + ln] = c[v] + bias;
    }
}

// ---------------------------------------------------------------------------
// Phase 2: sequential LSTM scan (single workgroup, 13 waves / 416 threads).
// Thread t < 400 owns row t of W_hh in registers; h, c, gates live in LDS.
// Final FC fused at the end (threads 0..19).
// ---------------------------------------------------------------------------
__global__ void __launch_bounds__(416, 1) lstm_scan_kernel(
    const float* __restrict__ xproj,
    const float* __restrict__ W_hh,
    const float* __restrict__ fc_w,
    const float* __restrict__ fc_b,
    float*       __restrict__ out)
{
    __shared__ __align__(16) float h_s[HID];
    __shared__ __align__(16) float c_s[HID];
    __shared__ __align__(16) float g_s[G4];

    const int t = threadIdx.x;

    // Preload this thread's W_hh row into registers (100 VGPRs/lane).
    float w[HID];
    if (t < G4) {
        const float* wr = W_hh + (long)t * HID;
        #pragma unroll
        for (int k = 0; k < HID; ++k) w[k] = wr[k];
    }
    if (t < HID) { h_s[t] = 0.0f; c_s[t] = 0.0f; }
    __syncthreads();

    for (int step = 0; step < SEQL; ++step) {
        if (t < G4) {
            float acc = xproj[(long)step * G4 + t];
            if (step + 1 < SEQL)
                __builtin_prefetch(&xproj[(long)(step + 1) * G4 + t], 0, 1);
            // dot(W_hh[t,:], h) with broadcast float4 LDS reads
            const float4* h4 = (const float4*)h_s;
            #pragma unroll
            for (int k = 0; k < HID / 4; ++k) {
                const float4 hv = h4[k];
                acc = fmaf(w[4 * k + 0], hv.x, acc);
                acc = fmaf(w[4 * k + 1], hv.y, acc);
                acc = fmaf(w[4 * k + 2], hv.z, acc);
                acc = fmaf(w[4 * k + 3], hv.w, acc);
            }
            g_s[t] = acc;
        }
        __syncthreads();
        if (t < HID) {
            const float i  = 1.0f / (1.0f + __expf(-g_s[t]));
            const float f  = 1.0f / (1.0f + __expf(-g_s[HID + t]));
            const float gg = tanhf(g_s[2 * HID + t]);
            const float o  = 1.0f / (1.0f + __expf(-g_s[3 * HID + t]));
            const float cn = f * c_s[t] + i * gg;
            c_s[t] = cn;
            h_s[t] = o * tanhf(cn);
        }
        __syncthreads();
    }

    // Final FC: out = h_last @ fc_w^T + fc_b  (20 outputs)
    if (t < NCLS) {
        float acc = fc_b[t];
        const float* fr = fc_w + (long)t * HID;
        #pragma unroll
        for (int k = 0; k < HID; ++k) acc = fmaf(fr[k], h_s[k], acc);
        out[t] = acc;
    }
}

// ---------------------------------------------------------------------------
extern "C" void kernel_launch(void* const* d_in, const int* in_sizes, int n_in,
                              void* d_out, int out_size, void* d_ws, size_t ws_size,
                              hipStream_t stream)
{
    const int*   seq = (const int*)  d_in[0];
    const float* emb = (const float*)d_in[1];
    const float* Wih = (const float*)d_in[2];
    const float* Whh = (const float*)d_in[3];
    const float* bih = (const float*)d_in[4];
    const float* bhh = (const float*)d_in[5];
    const float* fcw = (const float*)d_in[6];
    const float* fcb = (const float*)d_in[7];
    float* out = (float*)d_out;

    float* xproj = (float*)d_ws;   // SEQL * G4 floats = 13.1 MB scratch

    dim3 g1(SEQL / 16, G4 / 16);   // 512 x 25 tiles, one wave each
    xproj_wmma_kernel<<<g1, 32, 0, stream>>>(seq, emb, Wih, bih, bhh, xproj);
    lstm_scan_kernel<<<1, 416, 0, stream>>>(xproj, Whh, fcw, fcb, out);
}